// PSFE_41678362640403
// MI455X (gfx1250) — compile-verified
//
#include <hip/hip_runtime.h>

#define KNN  20
#define BB   4
#define NN   4096
#define PT   64
#define OUTC 256

typedef __attribute__((ext_vector_type(2))) float v2f;
typedef __attribute__((ext_vector_type(4))) float v4f;
typedef __attribute__((ext_vector_type(8))) float v8f;

#define WAIT_ASYNC() asm volatile("s_wait_asynccnt 0x0" ::: "memory")

// Async global -> LDS copy of 16 bytes: IOFFSET applies to both LDS dest and
// global source (ISA 15.18 GLOBAL_LOAD_ASYNC_TO_LDS pseudocode), so a single
// base pair + immediate offset walks a contiguous copy. Tracked by ASYNCcnt.
#define ASYNC_CP16(ldsa, gaddr, imm_off)                                    \
    asm volatile("global_load_async_to_lds_b128 %0, %1, off offset:%2"      \
                 :: "v"(ldsa), "v"(gaddr), "i"(imm_off) : "memory")

// ---------------------------------------------------------------------------
// KNN: one wave32 per query point. Lexicographic (dist, index) order matches
// jax.lax.top_k(-(d2+eps)) tie-breaking (smaller distance, then smaller idx).
// ---------------------------------------------------------------------------
__device__ __forceinline__ bool lexless(float d0, int i0, float d1, int i1) {
    return (d0 < d1) || (d0 == d1 && i0 < i1);
}

__global__ void __launch_bounds__(256) psfe_knn_kernel(const float* __restrict__ x,
                                                       int* __restrict__ idx) {
    const int wave = threadIdx.x >> 5;
    const int lane = threadIdx.x & 31;
    const int row  = blockIdx.x * 8 + wave;     // 0 .. B*N-1
    const int b    = row >> 12;                 // / 4096
    const int n    = row & (NN - 1);

    const float* xb = x + (size_t)b * NN * 3;
    const float qx = xb[n * 3 + 0];
    const float qy = xb[n * 3 + 1];
    const float qz = xb[n * 3 + 2];
    const float sqn = qx * qx + qy * qy + qz * qz;

    float bd[KNN];
    int   bi[KNN];
#pragma unroll
    for (int t = 0; t < KNN; ++t) { bd[t] = 3.4e38f; bi[t] = 0x7fffffff; }

    for (int j = lane; j < NN; j += 32) {
        float px = xb[j * 3 + 0];
        float py = xb[j * 3 + 1];
        float pz = xb[j * 3 + 2];
        float sqj = px * px + py * py + pz * pz;
        float d = sqn + sqj - 2.0f * (qx * px + qy * py + qz * pz);
        if (lexless(d, j, bd[KNN - 1], bi[KNN - 1])) {
            float cd = d; int ci = j;
#pragma unroll
            for (int t = 0; t < KNN; ++t) {
                bool c = lexless(cd, ci, bd[t], bi[t]);
                float td = bd[t]; int ti = bi[t];
                if (c) { bd[t] = cd; bi[t] = ci; cd = td; ci = ti; }
            }
        }
    }

    // 20 rounds of wave32 arg-min merge (each lane's list is sorted ascending)
    int* orow = idx + (size_t)row * KNN;
#pragma unroll 1
    for (int r = 0; r < KNN; ++r) {
        float d = bd[0]; int i = bi[0]; int wl = lane;
#pragma unroll
        for (int off = 16; off > 0; off >>= 1) {
            float od = __shfl_xor(d, off, 32);
            int   oi = __shfl_xor(i, off, 32);
            int   ol = __shfl_xor(wl, off, 32);
            if (lexless(od, oi, d, i)) { d = od; i = oi; wl = ol; }
        }
        if (lane == 0) orow[r] = i;
        if (wl == lane) {
#pragma unroll
            for (int t = 0; t < KNN - 1; ++t) { bd[t] = bd[t + 1]; bi[t] = bi[t + 1]; }
            bd[KNN - 1] = 3.4e38f; bi[KNN - 1] = 0x7fffffff;
        }
    }
}

// ---------------------------------------------------------------------------
// Conv1: (B,N,3) -> (B,N,64) point-major; trivial fan-in, plain VALU.
// ---------------------------------------------------------------------------
__global__ void __launch_bounds__(256) psfe_conv1_kernel(const float* __restrict__ x,
                                                         const float* __restrict__ W1,
                                                         const float* __restrict__ b1,
                                                         float* __restrict__ f1) {
    int pt = blockIdx.x * blockDim.x + threadIdx.x;     // 0 .. B*N-1
    if (pt >= BB * NN) return;
    float px = x[pt * 3 + 0];
    float py = x[pt * 3 + 1];
    float pz = x[pt * 3 + 2];
    float* o = f1 + (size_t)pt * PT;
#pragma unroll 8
    for (int oc = 0; oc < PT; ++oc) {
        o[oc] = b1[oc] + W1[oc * 3 + 0] * px + W1[oc * 3 + 1] * py + W1[oc * 3 + 2] * pz;
    }
}

// ---------------------------------------------------------------------------
// Dense 1x1 conv via fp32 WMMA: D(16 o x 16 pts) += A(W 16x4) x B(feat 4x16).
// fin/fout are point-major (B,N,C). One wave per (b, n-tile, o-tile).
// ---------------------------------------------------------------------------
template <int CIN, int COUT>
__global__ void __launch_bounds__(256) psfe_dense_wmma_kernel(const float* __restrict__ fin,
                                                              const float* __restrict__ W,
                                                              const float* __restrict__ bias,
                                                              float* __restrict__ fout) {
    const int wave = threadIdx.x >> 5;
    const int lane = threadIdx.x & 31;
    int gw = blockIdx.x * 8 + wave;
    constexpr int OT = COUT / 16;
    const int ot = gw % OT; gw /= OT;
    const int nt = gw % (NN / 16);
    const int b  = gw / (NN / 16);
    const int n0 = nt * 16;
    const int o0 = ot * 16;
    const int m  = lane & 15;
    const int hi = lane >> 4;

    v8f acc;
#pragma unroll
    for (int r = 0; r < 8; ++r) acc[r] = bias[o0 + 8 * hi + r];

    const float* wrow = W + (size_t)(o0 + m) * CIN + 2 * hi;
    const float* frow = fin + ((size_t)b * NN + n0 + m) * CIN + 2 * hi;
#pragma unroll 4
    for (int k0 = 0; k0 < CIN; k0 += 4) {
        v2f a  = *(const v2f*)(wrow + k0);
        v2f bm = *(const v2f*)(frow + k0);
        acc = __builtin_amdgcn_wmma_f32_16x16x4_f32(false, a, false, bm,
                                                    (short)0, acc, false, false);
    }

    // D[r]: channel = o0 + 8*hi + r, point = n0 + m -> 8 contiguous floats
    float* orow = fout + ((size_t)b * NN + n0 + m) * COUT + o0 + 8 * hi;
#pragma unroll
    for (int r = 0; r < 8; ++r) orow[r] = acc[r];
}

// ---------------------------------------------------------------------------
// Fused neighbor-gather + GEMM (the heavy layers).
// One block = one (b, 32-point tile). Async global->LDS gather of 4 neighbors
// per chunk, double-buffered (issue kc+1 while WMMA-ing kc). Each of 8 waves
// owns two 16-channel o-tiles x two 16-point n-tiles (weight loads reused).
// ---------------------------------------------------------------------------
template <int CIN>
__global__ void __launch_bounds__(256) psfe_gather_wmma_kernel(const float* __restrict__ fin,
                                                               const int* __restrict__ idx,
                                                               const float* __restrict__ W,
                                                               const float* __restrict__ bias,
                                                               float* __restrict__ out,
                                                               int chBase) {
    constexpr int KW  = KNN * CIN;      // 1280 / 2560 full kk reduction
    constexpr int CH  = 4 * CIN;        // kk-width of one chunk (4 neighbors)
    constexpr int STR = CH + 4;         // padded stride: 260/516 % 64 == 4 (bank-safe)
    constexpr int NPT = 32;             // points per block
    extern __shared__ float smem[];     // [2][NPT][STR]

    const int wave = threadIdx.x >> 5;
    const int lane = threadIdx.x & 31;
    const int nt = blockIdx.x % (NN / NPT);
    const int b  = blockIdx.x / (NN / NPT);
    const int n0 = nt * NPT;
    const int m  = lane & 15;
    const int hi = lane >> 4;
    const int o0 = wave * 32;           // this wave's two o-tiles: o0, o0+16

    v8f a00, a01, a10, a11;             // acc[o-tile][n-tile]
#pragma unroll
    for (int r = 0; r < 8; ++r) {
        float b0 = bias[o0 + 8 * hi + r];
        float b1 = bias[o0 + 16 + 8 * hi + r];
        a00[r] = b0; a01[r] = b0;
        a10[r] = b1; a11[r] = b1;
    }

    // gather assignment: 128 (point, neighbor) pairs, 2 threads per pair
    const int pairId = threadIdx.x >> 1;    // 0..127
    const int sub    = threadIdx.x & 1;     // 0..1
    const int gp     = pairId >> 2;         // point in tile 0..31
    const int gk     = pairId & 3;          // neighbor within chunk 0..3
    const int c0     = sub * (CIN / 2);

    const int* idxrow = idx + ((size_t)b * NN + n0 + gp) * KNN;
    const float* w0base = W + (size_t)(o0 + m) * KW + 2 * hi;
    const float* w1base = W + (size_t)(o0 + 16 + m) * KW + 2 * hi;

    // prologue: async-gather chunk 0 into buffer 0
    {
        int nb = idxrow[0 * 4 + gk];
        const float* src = fin + ((size_t)b * NN + nb) * CIN + c0;
        unsigned ldsa = (unsigned)(uintptr_t)(smem + gp * STR + gk * CIN + c0);
#pragma unroll
        for (int c = 0; c < CIN / 2; c += 4) ASYNC_CP16(ldsa, src, c * 4);
    }
    WAIT_ASYNC();
    __syncthreads();

#pragma unroll 1
    for (int kc = 0; kc < 5; ++kc) {
        const int buf = kc & 1;
        // ---- prefetch next chunk into the other buffer (async, no VGPR trip)
        if (kc < 4) {
            int nb = idxrow[(kc + 1) * 4 + gk];
            const float* src = fin + ((size_t)b * NN + nb) * CIN + c0;
            unsigned ldsa = (unsigned)(uintptr_t)
                (smem + (buf ^ 1) * NPT * STR + gp * STR + gk * CIN + c0);
#pragma unroll
            for (int c = 0; c < CIN / 2; c += 4) ASYNC_CP16(ldsa, src, c * 4);
        }

        // ---- compute current chunk: CIN k-steps of 4, 4 WMMAs per step
        const float* brow0 = smem + buf * NPT * STR + m * STR + 2 * hi;
        const float* brow1 = brow0 + 16 * STR;
        const float* w0 = w0base + kc * CH;
        const float* w1 = w1base + kc * CH;
#pragma unroll 8
        for (int s = 0; s < CIN; ++s) {
            v2f bm0 = *(const v2f*)(brow0 + 4 * s);
            v2f bm1 = *(const v2f*)(brow1 + 4 * s);
            v2f wa0 = *(const v2f*)(w0 + 4 * s);
            v2f wa1 = *(const v2f*)(w1 + 4 * s);
            a00 = __builtin_amdgcn_wmma_f32_16x16x4_f32(false, wa0, false, bm0,
                                                        (short)0, a00, false, false);
            a01 = __builtin_amdgcn_wmma_f32_16x16x4_f32(false, wa0, false, bm1,
                                                        (short)0, a01, false, false);
            a10 = __builtin_amdgcn_wmma_f32_16x16x4_f32(false, wa1, false, bm0,
                                                        (short)0, a10, false, false);
            a11 = __builtin_amdgcn_wmma_f32_16x16x4_f32(false, wa1, false, bm1,
                                                        (short)0, a11, false, false);
        }

        WAIT_ASYNC();       // own wave's async gathers landed in LDS
        __syncthreads();    // everyone's gathers visible; buffer reads done
    }

    // ---- store: out[b][chBase + ch][n] (channel-major final output)
#pragma unroll
    for (int r = 0; r < 8; ++r) {
        int ch0 = chBase + o0 + 8 * hi + r;
        size_t base0 = ((size_t)b * (4 * OUTC) + ch0) * NN;
        size_t base1 = ((size_t)b * (4 * OUTC) + ch0 + 16) * NN;
        out[base0 + n0 + m]      = a00[r];
        out[base0 + n0 + 16 + m] = a01[r];
        out[base1 + n0 + m]      = a10[r];
        out[base1 + n0 + 16 + m] = a11[r];
    }
}

// ---------------------------------------------------------------------------
// Launch pipeline. Inputs (setup_inputs order):
// 0:x 1:W1 2:b1 3:W2 4:b2 5:W3 6:b3 7:W4 8:b4
// 9:WL2 10:bL2 11:WL3 12:bL3 13:WL4 14:bL4 15:WL5 16:bL5
// ---------------------------------------------------------------------------
extern "C" void kernel_launch(void* const* d_in, const int* in_sizes, int n_in,
                              void* d_out, int out_size, void* d_ws, size_t ws_size,
                              hipStream_t stream) {
    (void)in_sizes; (void)n_in; (void)out_size; (void)ws_size;

    const float* x   = (const float*)d_in[0];
    const float* W1  = (const float*)d_in[1];  const float* b1  = (const float*)d_in[2];
    const float* W2  = (const float*)d_in[3];  const float* b2  = (const float*)d_in[4];
    const float* W3  = (const float*)d_in[5];  const float* b3  = (const float*)d_in[6];
    const float* W4  = (const float*)d_in[7];  const float* b4  = (const float*)d_in[8];
    const float* WL2 = (const float*)d_in[9];  const float* bL2 = (const float*)d_in[10];
    const float* WL3 = (const float*)d_in[11]; const float* bL3 = (const float*)d_in[12];
    const float* WL4 = (const float*)d_in[13]; const float* bL4 = (const float*)d_in[14];
    const float* WL5 = (const float*)d_in[15]; const float* bL5 = (const float*)d_in[16];
    float* out = (float*)d_out;

    char* ws = (char*)d_ws;
    int*   idx = (int*)ws;                                   // B*N*20 ints (1.25 MB)
    float* f1  = (float*)(ws + 1310720);                     // (B,N,64)
    float* f2  = f1 + (size_t)BB * NN * 64;                  // (B,N,64)
    float* f3  = f2 + (size_t)BB * NN * 64;                  // (B,N,64)
    float* f4  = f3 + (size_t)BB * NN * 64;                  // (B,N,128)

    // 1) KNN indices
    psfe_knn_kernel<<<(BB * NN) / 8, 256, 0, stream>>>(x, idx);

    // 2) conv1: 3 -> 64
    psfe_conv1_kernel<<<(BB * NN) / 256, 256, 0, stream>>>(x, W1, b1, f1);

    // 3) dense chain: 64->64, 64->64, 64->128
    psfe_dense_wmma_kernel<64, 64><<<(BB * (NN / 16) * 4) / 8, 256, 0, stream>>>(f1, W2, b2, f2);
    psfe_dense_wmma_kernel<64, 64><<<(BB * (NN / 16) * 4) / 8, 256, 0, stream>>>(f2, W3, b3, f3);
    psfe_dense_wmma_kernel<64, 128><<<(BB * (NN / 16) * 8) / 8, 256, 0, stream>>>(f3, W4, b4, f4);

    // 4) gather+GEMM layers -> output channel blocks (double-buffered LDS)
    const size_t lds64  = 2u * 32u * (4 * 64  + 4) * sizeof(float);   //  66,560 B
    const size_t lds128 = 2u * 32u * (4 * 128 + 4) * sizeof(float);   // 132,096 B
    psfe_gather_wmma_kernel<64> <<<BB * (NN / 32), 256, lds64,  stream>>>(f1, idx, WL2, bL2, out, 0);
    psfe_gather_wmma_kernel<64> <<<BB * (NN / 32), 256, lds64,  stream>>>(f2, idx, WL3, bL3, out, 256);
    psfe_gather_wmma_kernel<64> <<<BB * (NN / 32), 256, lds64,  stream>>>(f3, idx, WL4, bL4, out, 512);
    psfe_gather_wmma_kernel<128><<<BB * (NN / 32), 256, lds128, stream>>>(f4, idx, WL5, bL5, out, 768);
}